// ContextualCentroidPerception_24567212933823
// MI455X (gfx1250) — compile-verified
//
#include <hip/hip_runtime.h>
#include <hip/hip_bf16.h>
#include <stdint.h>

#define BATCH 4
#define NPTS  65536
#define KSEL  4096
#define CIN   256
#define MID   128
#define NPROD (BATCH * KSEL)      // 16384 rows through the MLP
#define BN_EPS 1e-5f

typedef float v2f __attribute__((ext_vector_type(2)));
typedef float v8f __attribute__((ext_vector_type(8)));

// ---------------------------------------------------------------------------
// K0: scores -> order-preserving uint keys (larger key == larger score)
// ---------------------------------------------------------------------------
__global__ __launch_bounds__(256) void ccp_score_kernel(
    const float* __restrict__ cls_preds, uint32_t* __restrict__ keys)
{
    int i = blockIdx.x * 256 + threadIdx.x;          // i in [0, B*N)
    if (i >= BATCH * NPTS) return;
    float a = cls_preds[(size_t)i * 3 + 0];
    float b = cls_preds[(size_t)i * 3 + 1];
    float c = cls_preds[(size_t)i * 3 + 2];
    float s = fmaxf(a, fmaxf(b, c));
    uint32_t u = __float_as_uint(s);
    keys[i] = (u & 0x80000000u) ? ~u : (u | 0x80000000u);
}

// ---------------------------------------------------------------------------
// K1: per-batch exact top-K (value desc, index asc on ties) via radix-select
//     + bitonic sort of the K survivors. One 1024-thread block per batch.
// ---------------------------------------------------------------------------
__global__ __launch_bounds__(1024) void ccp_topk_kernel(
    const uint32_t* __restrict__ keys, const float* __restrict__ points,
    int* __restrict__ idxOut, float* __restrict__ outOrigins)
{
    const int b = blockIdx.x;
    const int t = threadIdx.x;
    const uint32_t* bk = keys + (size_t)b * NPTS;

    __shared__ unsigned long long sh[KSEL];          // 32 KB
    __shared__ uint32_t hist[256];
    __shared__ uint32_t eqPre[1025];
    __shared__ uint32_t s_prefix, s_remaining, s_cnt;

    if (t == 0) { s_prefix = 0u; s_remaining = KSEL; s_cnt = 0u; }
    __syncthreads();

    // 4-pass MSB radix select: find threshold key T (the K-th largest)
    for (int p = 0; p < 4; ++p) {
        const int shift = 24 - 8 * p;
        if (t < 256) hist[t] = 0u;
        __syncthreads();
        uint32_t pref = s_prefix;
        const int base = t * 64;                      // contiguous chunk per thread
        for (int i = 0; i < 64; ++i) {
            uint32_t kv = bk[base + i];
            if ((((unsigned long long)(kv ^ pref)) >> (shift + 8)) == 0ull)
                atomicAdd(&hist[(kv >> shift) & 255u], 1u);
        }
        __syncthreads();
        if (t == 0) {
            uint32_t rem = s_remaining, cum = 0u;
            int bin = 0;
            for (int q = 255; q >= 0; --q) {
                uint32_t c = hist[q];
                if (cum + c >= rem) { bin = q; s_remaining = rem - cum; break; }
                cum += c;
            }
            s_prefix = pref | ((uint32_t)bin << shift);
        }
        __syncthreads();
    }
    const uint32_t T = s_prefix;
    const uint32_t need = s_remaining;               // # of ties at T to keep

    // stable tie ranking: chunks are index-ordered, prefix-sum per-thread counts
    const int base = t * 64;
    uint32_t eq = 0;
    for (int i = 0; i < 64; ++i) {
        uint32_t kv = bk[base + i];
        if (kv > T) {
            uint32_t pos = atomicAdd(&s_cnt, 1u);
            sh[pos] = ((unsigned long long)(~kv) << 32) | (uint32_t)(base + i);
        } else if (kv == T) {
            ++eq;
        }
    }
    eqPre[t + 1] = eq;
    __syncthreads();
    if (t == 0) { eqPre[0] = 0; for (int i = 1; i <= 1024; ++i) eqPre[i] += eqPre[i - 1]; }
    __syncthreads();
    uint32_t r = eqPre[t];
    for (int i = 0; i < 64 && r < need; ++i) {
        uint32_t kv = bk[base + i];
        if (kv == T) {
            uint32_t pos = atomicAdd(&s_cnt, 1u);
            sh[pos] = ((unsigned long long)(~kv) << 32) | (uint32_t)(base + i);
            ++r;
        }
    }
    __syncthreads();

    // bitonic sort ascending: smallest packed key == highest score, ties by idx
    for (unsigned k2 = 2; k2 <= KSEL; k2 <<= 1) {
        for (unsigned j = k2 >> 1; j > 0; j >>= 1) {
            for (int i = t; i < KSEL; i += 1024) {
                int l = i ^ (int)j;
                if (l > i) {
                    bool up = ((i & k2) == 0);
                    unsigned long long A = sh[i], Bv = sh[l];
                    if ((A > Bv) == up) { sh[i] = Bv; sh[l] = A; }
                }
            }
            __syncthreads();
        }
    }

    // emit idx + gathered origins
    for (int i = t; i < KSEL; i += 1024) {
        int n = (int)(sh[i] & 0xffffffffull);
        idxOut[b * KSEL + i] = n;
        size_t src = ((size_t)b * NPTS + n) * 3;
        size_t dst = ((size_t)b * KSEL + i) * 3;
        outOrigins[dst + 0] = points[src + 0];
        outOrigins[dst + 1] = points[src + 1];
        outOrigins[dst + 2] = points[src + 2];
    }
}

// ---------------------------------------------------------------------------
// K2: fused gather + GEMM1 via fp32 WMMA (V_WMMA_F32_16X16X4_F32).
//     Block: 256 threads = 8 waves, tile = 32 points x 128 outputs.
//     Wave w: rows 16*(w&1).., cols [ (w>>1)*32 , +16 ] (two 16x16 tiles).
// ---------------------------------------------------------------------------
#define FSTRIDE 258   // LDS row stride (floats) -> conflict-free A fragment loads

__global__ __launch_bounds__(256) void ccp_gemm1_kernel(
    const float* __restrict__ features, const float* __restrict__ w1,
    const int* __restrict__ idxBuf, float* __restrict__ h)
{
    __shared__ float sFeat[32 * FSTRIDE];            // ~33 KB
    __shared__ int   sIdx[32];

    const int p0 = blockIdx.x * 32;                  // first point row of tile
    const int b  = p0 / KSEL;
    const int t  = threadIdx.x;

    if (t < 32) sIdx[t] = idxBuf[p0 + t];
    __syncthreads();

    // gather: thread t == channel t; 32 feature vectors, stride-N in memory
    for (int i = 0; i < 32; ++i) {
        int n = sIdx[i];
        sFeat[i * FSTRIDE + t] =
            features[((size_t)b * CIN + t) * NPTS + (size_t)n];
    }
    __syncthreads();

    const int wid  = t >> 5;
    const int lane = t & 31;
    const int mBase = (wid & 1) * 16;
    const int n0 = (wid >> 1) * 32;
    const int n1 = n0 + 16;

    const int mrow = mBase + (lane & 15);
    const int koff = (lane >> 4) * 2;                // lanes 16..31 hold K+2,K+3
    const float* ap  = &sFeat[mrow * FSTRIDE + koff];
    const float* bp0 = w1 + (size_t)(n0 + (lane & 15)) * CIN + koff;
    const float* bp1 = w1 + (size_t)(n1 + (lane & 15)) * CIN + koff;

    v8f acc0 = {};
    v8f acc1 = {};
    for (int k = 0; k < CIN; k += 4) {
        v2f a  = { ap[k],  ap[k + 1]  };
        v2f b0 = { bp0[k], bp0[k + 1] };
        v2f b1 = { bp1[k], bp1[k + 1] };
        acc0 = __builtin_amdgcn_wmma_f32_16x16x4_f32(false, a, false, b0,
                                                     (short)0, acc0, false, false);
        acc1 = __builtin_amdgcn_wmma_f32_16x16x4_f32(false, a, false, b1,
                                                     (short)0, acc1, false, false);
    }

    // C/D layout: VGPR v -> M = v + 8*(lane>=16); N = lane&15
    const int col0 = n0 + (lane & 15);
    const int col1 = n1 + (lane & 15);
    const int rowB = p0 + mBase + ((lane >> 4) << 3);
    #pragma unroll
    for (int v = 0; v < 8; ++v) {
        h[(size_t)(rowB + v) * MID + col0] = acc0[v];
        h[(size_t)(rowB + v) * MID + col1] = acc1[v];
    }
}

// ---------------------------------------------------------------------------
// K3a: per-channel partial sums (deterministic, no float atomics)
// ---------------------------------------------------------------------------
__global__ __launch_bounds__(128) void ccp_bnpart_kernel(
    const float* __restrict__ h, float* __restrict__ pSum, float* __restrict__ pSq)
{
    const int c = threadIdx.x;
    const int j = blockIdx.x;                         // 64 blocks x 256 rows
    float s = 0.f, sq = 0.f;
    const int base = j * 256;
    for (int r = 0; r < 256; ++r) {
        float v = h[(size_t)(base + r) * MID + c];
        s += v; sq += v * v;
    }
    pSum[j * MID + c] = s;
    pSq [j * MID + c] = sq;
}

// K3b: finalize BN scale/shift
__global__ __launch_bounds__(128) void ccp_bnfin_kernel(
    const float* __restrict__ pSum, const float* __restrict__ pSq,
    const float* __restrict__ gamma, const float* __restrict__ beta,
    float* __restrict__ scale, float* __restrict__ shift)
{
    const int c = threadIdx.x;
    float s = 0.f, sq = 0.f;
    for (int j = 0; j < 64; ++j) { s += pSum[j * MID + c]; sq += pSq[j * MID + c]; }
    const float inv = 1.0f / (float)NPROD;
    float mean = s * inv;
    float var  = sq * inv - mean * mean;
    float sc   = gamma[c] * rsqrtf(var + BN_EPS);
    scale[c] = sc;
    shift[c] = beta[c] - mean * sc;
}

// ---------------------------------------------------------------------------
// K4: BN+ReLU + GEMM2(128->3) + clamp + add origins. One wave per point.
// ---------------------------------------------------------------------------
__global__ __launch_bounds__(256) void ccp_final_kernel(
    const float* __restrict__ h, const float* __restrict__ scale,
    const float* __restrict__ shift, const float* __restrict__ w2,
    const float* __restrict__ lim, float* __restrict__ out)
{
    const int lane = threadIdx.x & 31;
    const int wid  = threadIdx.x >> 5;
    const int p    = blockIdx.x * 8 + wid;            // point row in [0, NPROD)

    float a0 = 0.f, a1 = 0.f, a2 = 0.f;
    #pragma unroll
    for (int q = 0; q < 4; ++q) {
        int c = lane + q * 32;
        float v = h[(size_t)p * MID + c];
        v = fmaxf(v * scale[c] + shift[c], 0.f);      // BN + ReLU
        a0 += v * w2[0 * MID + c];
        a1 += v * w2[1 * MID + c];
        a2 += v * w2[2 * MID + c];
    }
    #pragma unroll
    for (int off = 16; off > 0; off >>= 1) {
        a0 += __shfl_down(a0, off, 32);
        a1 += __shfl_down(a1, off, 32);
        a2 += __shfl_down(a2, off, 32);
    }
    if (lane == 0) {
        float* preds   = out;
        float* origins = out + (size_t)NPROD * 3;
        float* offsets = out + (size_t)NPROD * 6;
        float o[3] = { a0, a1, a2 };
        #pragma unroll
        for (int d = 0; d < 3; ++d) {
            float L  = lim[d];
            float cl = fminf(fmaxf(o[d], -L), L);
            float org = origins[(size_t)p * 3 + d];
            offsets[(size_t)p * 3 + d] = o[d];
            preds  [(size_t)p * 3 + d] = org + cl;
        }
    }
}

// ---------------------------------------------------------------------------
extern "C" void kernel_launch(void* const* d_in, const int* in_sizes, int n_in,
                              void* d_out, int out_size, void* d_ws, size_t ws_size,
                              hipStream_t stream)
{
    const float* points   = (const float*)d_in[0];   // (B,N,3)
    const float* features = (const float*)d_in[1];   // (B,C,N)
    const float* cls      = (const float*)d_in[2];   // (B,N,3)
    const float* w1       = (const float*)d_in[3];   // (MID,C)
    const float* gamma    = (const float*)d_in[4];   // (MID)
    const float* beta     = (const float*)d_in[5];   // (MID)
    const float* w2       = (const float*)d_in[6];   // (3,MID)
    const float* lim      = (const float*)d_in[7];   // (1,1,3)
    float* out = (float*)d_out;

    char* ws = (char*)d_ws;
    uint32_t* keys   = (uint32_t*)(ws);                               // B*N u32
    int*      idxBuf = (int*)     (ws + (size_t)BATCH * NPTS * 4);    // B*K i32
    float*    h      = (float*)   (ws + (size_t)BATCH * NPTS * 4
                                      + (size_t)BATCH * KSEL * 4);    // NPROD*MID f32
    float*    pSum   = h + (size_t)NPROD * MID;                       // 64*MID
    float*    pSq    = pSum + 64 * MID;                               // 64*MID
    float*    scale  = pSq  + 64 * MID;                               // MID
    float*    shift  = scale + MID;                                   // MID

    float* outOrigins = out + (size_t)NPROD * 3;

    ccp_score_kernel<<<(BATCH * NPTS) / 256, 256, 0, stream>>>(cls, keys);
    ccp_topk_kernel<<<BATCH, 1024, 0, stream>>>(keys, points, idxBuf, outOrigins);
    ccp_gemm1_kernel<<<NPROD / 32, 256, 0, stream>>>(features, w1, idxBuf, h);
    ccp_bnpart_kernel<<<64, 128, 0, stream>>>(h, pSum, pSq);
    ccp_bnfin_kernel<<<1, 128, 0, stream>>>(pSum, pSq, gamma, beta, scale, shift);
    ccp_final_kernel<<<NPROD / 8, 256, 0, stream>>>(h, scale, shift, w2, lim, out);
}